// MaIR_82231443849568
// MI455X (gfx1250) — compile-verified
//
#include <hip/hip_runtime.h>
#include <math.h>

// Problem constants (from setup_inputs)
#define BDIM 4
#define Hh   64
#define Ww   64
#define DMO  180      // model dim
#define Din  360      // d_inner
#define Kdir 4
#define Nst  16       // d_state
#define RNK  12       // dt_rank
#define Lseq 4096
#define BL   (BDIM * Lseq)   // 16384
#define E2   (2 * Din)       // 720
#define RBC  44              // rank + 2N

typedef float v2f __attribute__((ext_vector_type(2)));
typedef float v8f __attribute__((ext_vector_type(8)));
typedef unsigned int u32x4 __attribute__((ext_vector_type(4)));
typedef int i32x4 __attribute__((ext_vector_type(4)));
typedef int i32x8 __attribute__((ext_vector_type(8)));

__device__ __forceinline__ v2f ld2(const float* p) {
  return *reinterpret_cast<const v2f*>(p);
}

// fp32 WMMA: D = A(16x4) * B(4x16) + C, full fp32 precision on CDNA5
__device__ __forceinline__ v8f wmma_f32(v2f a, v2f b, v8f c) {
  return __builtin_amdgcn_wmma_f32_16x16x4_f32(
      /*neg_a=*/false, a, /*neg_b=*/false, b,
      /*c_mod=*/(short)0, c, /*reuse_a=*/false, /*reuse_b=*/false);
}

// ---------------------------------------------------------------------------
// Tensor Data Mover: 2-D f32 tile (rows x cols, row stride in elements) from
// global memory into LDS at byte offset lds_off. D# per CDNA5 ISA 8.3/8.4:
//  g0: count=1 | lds_addr | global_addr[56:0] | type=2
//  g1: data_size=4B, tensor_dim0=cols, tensor_dim1=rows,
//      tile_dim0=cols, tile_dim1=rows (tile_dim2=0), dim0_stride=stride
// ---------------------------------------------------------------------------
__device__ __forceinline__ void tdm_load_2d(unsigned lds_off, const float* gsrc,
                                            unsigned rows, unsigned cols,
                                            unsigned stride) {
  unsigned long long ga = (unsigned long long)(size_t)gsrc;
  u32x4 g0 = { 1u,                                   // count=1, no gather
               lds_off,                              // lds_addr (bytes)
               (unsigned)ga,                         // global_addr[31:0]
               ((unsigned)(ga >> 32) & 0x01FFFFFFu) | (2u << 30) }; // [56:32]|type=2
  i32x8 g1 = { (int)(2u << 16),                              // data_size = 4B
               (int)((cols & 0xFFFFu) << 16),                // tensor_dim0 lo16
               (int)(((cols >> 16) & 0xFFFFu) | ((rows & 0xFFFFu) << 16)),
               (int)(((rows >> 16) & 0xFFFFu) | ((cols & 0xFFFFu) << 16)), // tile_dim0
               (int)(rows & 0xFFFFu),                        // tile_dim1 (tile_dim2=0)
               (int)stride,                                  // tensor_dim0_stride lo32
               0, 0 };                                       // dim1_stride unused
  i32x4 z4 = {0, 0, 0, 0};
#if defined(__clang_major__) && __clang_major__ >= 23
  i32x8 z8 = {0, 0, 0, 0, 0, 0, 0, 0};
  __builtin_amdgcn_tensor_load_to_lds(g0, g1, z4, z4, z8, 0);
#else
  __builtin_amdgcn_tensor_load_to_lds(g0, g1, z4, z4, 0);
#endif
}

// ---------------------------------------------------------------------------
// K0: in_proj GEMM  xz[bl][e] = sum_c x[bl][c] * ipw[e][c]   (K=180, N=720)
// one wave per 16x48 tile: A fragment reused across 3 accumulators
// ---------------------------------------------------------------------------
__global__ void k0_in_proj(const float* __restrict__ x, const float* __restrict__ w,
                           float* __restrict__ xz) {
  const int lane = threadIdx.x & 31;
  const int wave = (blockIdx.x * blockDim.x + threadIdx.x) >> 5;
  const int NG = E2 / 48;                      // 15 groups of 48 columns
  const int mt = wave / NG, ng = wave % NG;
  const int m0 = mt * 16, n0 = ng * 48;
  const int rr = lane & 15, kh = (lane >> 4) * 2;
  const float* arow = x + (size_t)(m0 + rr) * DMO + kh;        // A[M][K]
  const float* b0 = w + (size_t)(n0 + rr) * DMO + kh;          // B[K][N] = W[N][K]
  const float* b1 = w + (size_t)(n0 + 16 + rr) * DMO + kh;
  const float* b2 = w + (size_t)(n0 + 32 + rr) * DMO + kh;
  v8f c0 = {}, c1 = {}, c2 = {};
  for (int kk = 0; kk < DMO; kk += 4) {
    v2f a = ld2(arow + kk);
    c0 = wmma_f32(a, ld2(b0 + kk), c0);
    c1 = wmma_f32(a, ld2(b1 + kk), c1);
    c2 = wmma_f32(a, ld2(b2 + kk), c2);
  }
  const int row0 = (lane >> 4) * 8;
  float* orow = xz + (size_t)m0 * E2 + n0 + rr;
#pragma unroll
  for (int v = 0; v < 8; ++v) {
    size_t o = (size_t)(row0 + v) * E2;
    orow[o] = c0[v];
    orow[o + 16] = c1[v];
    orow[o + 32] = c2[v];
  }
}

// ---------------------------------------------------------------------------
// K1: depthwise 3x3 conv (SAME, cross-correlation) + SiLU
// xz[..][0..359] is xin; output xf[b][l][d] (channel-contiguous)
// ---------------------------------------------------------------------------
__global__ void k1_conv_silu(const float* __restrict__ xz, const float* __restrict__ cw,
                             const float* __restrict__ cb, float* __restrict__ xf) {
  int idx = blockIdx.x * blockDim.x + threadIdx.x;
  if (idx >= BL * Din) return;
  int d = idx % Din;
  int bl = idx / Din;
  int b = bl / Lseq, l = bl % Lseq;
  int h = l >> 6, w = l & 63;
  float acc = cb[d];
#pragma unroll
  for (int i = 0; i < 3; ++i) {
    int hh = h + i - 1;
    if (hh < 0 || hh >= Hh) continue;
#pragma unroll
    for (int j = 0; j < 3; ++j) {
      int ww = w + j - 1;
      if (ww < 0 || ww >= Ww) continue;
      acc = fmaf(cw[d * 9 + i * 3 + j],
                 xz[(size_t)(b * Lseq + (hh << 6) + ww) * E2 + d], acc);
    }
  }
  xf[(size_t)bl * Din + d] = acc / (1.0f + __expf(-acc));   // SiLU
}

// ---------------------------------------------------------------------------
// K2: x_dbl projection per (b,k): xdbl[b][k][r][l] = sum_d xpw[k][r][d]*xf[b][l][d]
// The 44x360 weight block is staged once per workgroup into LDS via the
// Tensor Data Mover; each wave then computes a 44(x48 padded)x16 column slab
// reading A-fragments from LDS (ds_load_b64) and B from global.
// 32 blocks per (b,k); 8 waves/block -> 256 l-tiles per (b,k).
// ---------------------------------------------------------------------------
__global__ void k2_xdbl(const float* __restrict__ xf, const float* __restrict__ xpw,
                        float* __restrict__ xdbl) {
  __shared__ float wlds[RBC * Din];               // 44*360*4 = 63360 B
  const int bk = blockIdx.x >> 5;                 // blockIdx / 32
  const int b = bk >> 2, k = bk & 3;
  if ((threadIdx.x >> 5) == 0) {                  // wave-uniform: wave 0 only
    tdm_load_2d((unsigned)(size_t)(void*)wlds,
                xpw + (size_t)k * RBC * Din, RBC, Din, Din);
    __builtin_amdgcn_s_wait_tensorcnt(0);         // TDM complete for this wave
  }
  __syncthreads();                                // publish LDS to all waves

  const int lane = threadIdx.x & 31;
  const int lt = (blockIdx.x & 31) * 8 + (threadIdx.x >> 5);  // 0..255
  const int l0 = lt * 16;
  const int rr = lane & 15, kh = (lane >> 4) * 2;
  const bool a2v = (32 + rr) < RBC;
  const float* a0 = wlds + (size_t)rr * Din + kh;             // rows 0..15
  const float* a1 = wlds + (size_t)(16 + rr) * Din + kh;      // rows 16..31
  const float* a2 = wlds + (size_t)(a2v ? 32 + rr : 0) * Din + kh;
  const float* brow = xf + (size_t)(b * Lseq + l0 + rr) * Din + kh;
  const v2f zz = {0.f, 0.f};
  v8f c0 = {}, c1 = {}, c2 = {};
  for (int kk = 0; kk < Din; kk += 4) {
    v2f bv = ld2(brow + kk);
    c0 = wmma_f32(ld2(a0 + kk), bv, c0);
    c1 = wmma_f32(ld2(a1 + kk), bv, c1);
    c2 = wmma_f32(a2v ? ld2(a2 + kk) : zz, bv, c2);
  }
  const int row0 = (lane >> 4) * 8;
  float* outp = xdbl + (size_t)(bk * RBC) * Lseq + l0 + rr;
#pragma unroll
  for (int v = 0; v < 8; ++v) {
    outp[(size_t)(row0 + v) * Lseq] = c0[v];
    outp[(size_t)(16 + row0 + v) * Lseq] = c1[v];
    int r2 = 32 + row0 + v;
    if (r2 < RBC) outp[(size_t)r2 * Lseq] = c2[v];
  }
}

// ---------------------------------------------------------------------------
// K3: selective scan. One thread per (b,k,d); 16 states in registers.
// dt-projection (12 FMAs) fused in. Gather via mair_ids, scatter back raster.
// ---------------------------------------------------------------------------
__global__ void k3_scan(const float* __restrict__ xf, const float* __restrict__ xdbl,
                        const int* __restrict__ mair,
                        const float* __restrict__ dtw, const float* __restrict__ dtb,
                        const float* __restrict__ alog, const float* __restrict__ Dsk,
                        float* __restrict__ ys, float* __restrict__ mpool) {
  int idx = blockIdx.x * blockDim.x + threadIdx.x;
  if (idx >= BDIM * Kdir * Din) return;
  int d = idx % Din;
  int k = (idx / Din) % Kdir;
  int b = idx / (Din * Kdir);
  int kd = k * Din + d;

  float A[Nst], h[Nst];
#pragma unroll
  for (int n = 0; n < Nst; ++n) { A[n] = -__expf(alog[kd * Nst + n]); h[n] = 0.f; }
  float w[RNK];
#pragma unroll
  for (int r = 0; r < RNK; ++r) w[r] = dtw[kd * RNK + r];
  const float bias = dtb[kd];
  const float Dk = Dsk[kd];

  const float* xd  = xdbl + (size_t)((b * Kdir + k) * RBC) * Lseq;  // [r][l]
  const int*   ids = mair + k * Lseq;
  const float* xfb = xf + (size_t)b * Lseq * Din + d;
  float* ysb = ys + (size_t)(b * Kdir + k) * Lseq * Din + d;

  float ysum = 0.f;
  for (int l = 0; l < Lseq; ++l) {
    int m = ids[l];
    int mn = ids[(l + 1 < Lseq) ? (l + 1) : l];
    __builtin_prefetch(xfb + (size_t)mn * Din, 0, 1);   // global_prefetch_b8
    float u = xfb[(size_t)m * Din];
    float raw = bias;
#pragma unroll
    for (int r = 0; r < RNK; ++r) raw = fmaf(w[r], xd[(size_t)r * Lseq + m], raw);
    float dt = (raw > 20.f) ? raw : log1pf(__expf(raw));   // softplus
    float du = dt * u;
    float y = 0.f;
#pragma unroll
    for (int n = 0; n < Nst; ++n) {
      float Bn = xd[(size_t)(RNK + n) * Lseq + m];
      float Cn = xd[(size_t)(RNK + Nst + n) * Lseq + m];
      h[n] = fmaf(h[n], __expf(dt * A[n]), du * Bn);
      y = fmaf(h[n], Cn, y);
    }
    y = fmaf(Dk, u, y);                  // + D*u skip
    ysb[(size_t)m * Din] = y;            // scatter to raster order (inverse perm)
    ysum += y;
  }
  mpool[(b * Kdir + k) * Din + d] = ysum * (1.0f / Lseq);
}

// ---------------------------------------------------------------------------
// K4: ShuffleAttn gates  (tiny)
// ---------------------------------------------------------------------------
__global__ void k4_gates(const float* __restrict__ mpool, const float* __restrict__ gw,
                         const float* __restrict__ gb, float* __restrict__ gates) {
  int idx = blockIdx.x * blockDim.x + threadIdx.x;
  if (idx >= BDIM * Kdir * Din) return;
  int d = idx % Din;
  int j = (idx / Din) % Kdir;
  int b = idx / (Din * Kdir);
  float acc = gb[d * Kdir + j];
#pragma unroll
  for (int i = 0; i < Kdir; ++i)
    acc = fmaf(gw[(d * Kdir + j) * Kdir + i], mpool[(b * Kdir + i) * Din + d], acc);
  gates[(b * Kdir + j) * Din + d] = 1.0f / (1.0f + __expf(-acc));
}

// ---------------------------------------------------------------------------
// K5: gated sum over K + LayerNorm(d) + silu(z) gate  -> yact[b][l][d]
// one block per (b,l); LDS block reduction for mean/var
// ---------------------------------------------------------------------------
__global__ void k5_combine_ln(const float* __restrict__ ys, const float* __restrict__ gates,
                              const float* __restrict__ xz,
                              const float* __restrict__ onw, const float* __restrict__ onb,
                              float* __restrict__ yact) {
  __shared__ float s1[512];
  __shared__ float s2[512];
  int bl = blockIdx.x;
  int b = bl / Lseq, l = bl % Lseq;
  int d = threadIdx.x;
  float v = 0.f;
  if (d < Din) {
#pragma unroll
    for (int k = 0; k < Kdir; ++k)
      v = fmaf(ys[((size_t)(b * Kdir + k) * Lseq + l) * Din + d],
               gates[(b * Kdir + k) * Din + d], v);
  }
  s1[threadIdx.x] = (d < Din) ? v : 0.f;
  s2[threadIdx.x] = (d < Din) ? v * v : 0.f;
  __syncthreads();
  for (int s = 256; s > 0; s >>= 1) {
    if (threadIdx.x < s) {
      s1[threadIdx.x] += s1[threadIdx.x + s];
      s2[threadIdx.x] += s2[threadIdx.x + s];
    }
    __syncthreads();
  }
  float mu  = s1[0] * (1.0f / Din);
  float var = s2[0] * (1.0f / Din) - mu * mu;
  float inv = rsqrtf(var + 1e-5f);
  if (d < Din) {
    float yn = (v - mu) * inv * onw[d] + onb[d];
    float z = xz[(size_t)bl * E2 + Din + d];
    float sz = z / (1.0f + __expf(-z));
    yact[(size_t)bl * Din + d] = yn * sz;
  }
}

// ---------------------------------------------------------------------------
// K6: out_proj GEMM  out[bl][c] = sum_d yact[bl][d]*opw[c][d]  (N=180 pad 192)
// one wave per 16x64 tile group: A fragment reused across 4 accumulators
// ---------------------------------------------------------------------------
__global__ void k6_out_proj(const float* __restrict__ yact, const float* __restrict__ w,
                            float* __restrict__ out) {
  const int lane = threadIdx.x & 31;
  const int wave = (blockIdx.x * blockDim.x + threadIdx.x) >> 5;
  const int NG = 3;                             // 3 groups of 64 columns (pad 192)
  const int mt = wave / NG, ng = wave % NG;
  const int m0 = mt * 16, n0 = ng * 64;
  const int rr = lane & 15, kh = (lane >> 4) * 2;
  const float* arow = yact + (size_t)(m0 + rr) * Din + kh;
  const v2f zz = {0.f, 0.f};
  const float* bp[4];
  bool bv[4];
#pragma unroll
  for (int j = 0; j < 4; ++j) {
    int cb = n0 + j * 16 + rr;
    bv[j] = cb < DMO;
    bp[j] = w + (size_t)(bv[j] ? cb : 0) * Din + kh;
  }
  v8f c[4] = {};
  for (int kk = 0; kk < Din; kk += 4) {
    v2f a = ld2(arow + kk);
#pragma unroll
    for (int j = 0; j < 4; ++j)
      c[j] = wmma_f32(a, bv[j] ? ld2(bp[j] + kk) : zz, c[j]);
  }
  const int row0 = (lane >> 4) * 8;
#pragma unroll
  for (int j = 0; j < 4; ++j) {
    int cc = n0 + j * 16 + rr;
    if (cc < DMO) {
#pragma unroll
      for (int v = 0; v < 8; ++v)
        out[(size_t)(m0 + row0 + v) * DMO + cc] = c[j][v];
    }
  }
}

// ---------------------------------------------------------------------------
extern "C" void kernel_launch(void* const* d_in, const int* in_sizes, int n_in,
                              void* d_out, int out_size, void* d_ws, size_t ws_size,
                              hipStream_t stream) {
  const float* x    = (const float*)d_in[0];
  const int*   mair = (const int*)d_in[1];
  const float* ipw  = (const float*)d_in[2];
  const float* cw   = (const float*)d_in[3];
  const float* cb   = (const float*)d_in[4];
  const float* xpw  = (const float*)d_in[5];
  const float* dtw  = (const float*)d_in[6];
  const float* dtb  = (const float*)d_in[7];
  const float* alog = (const float*)d_in[8];
  const float* ds   = (const float*)d_in[9];
  const float* onw  = (const float*)d_in[10];
  const float* onb  = (const float*)d_in[11];
  const float* opw  = (const float*)d_in[12];
  const float* gw   = (const float*)d_in[13];
  const float* gb   = (const float*)d_in[14];
  float* out = (float*)d_out;

  float* ws = (float*)d_ws;
  size_t off = 0;
  float* xz    = ws + off; off += (size_t)BL * E2;                    // 11.8M f32
  float* xf    = ws + off; off += (size_t)BL * Din;                   // 5.9M
  float* xdbl  = ws + off; off += (size_t)BDIM * Kdir * RBC * Lseq;   // 2.9M
  float* ys    = ws + off; off += (size_t)BDIM * Kdir * Lseq * Din;   // 23.6M
  float* mpool = ws + off; off += (size_t)BDIM * Kdir * Din;
  float* gates = ws + off; off += (size_t)BDIM * Kdir * Din;
  float* yact  = ws + off; off += (size_t)BL * Din;                   // 5.9M

  // K0: 1024 M-tiles x 15 column-groups, 8 waves/block
  k0_in_proj<<<(1024 * 15) / 8, 256, 0, stream>>>(x, ipw, xz);
  // K1: depthwise conv + SiLU
  k1_conv_silu<<<(BL * Din + 255) / 256, 256, 0, stream>>>(xz, cw, cb, xf);
  // K2: 16 (b,k) x 32 blocks; TDM-staged weights in LDS
  k2_xdbl<<<16 * 32, 256, 0, stream>>>(xf, xpw, xdbl);
  // K3: selective scan, 5760 sequences
  k3_scan<<<(BDIM * Kdir * Din) / 128, 128, 0, stream>>>(xf, xdbl, mair, dtw, dtb,
                                                         alog, ds, ys, mpool);
  // K4: gates
  k4_gates<<<(BDIM * Kdir * Din + 255) / 256, 256, 0, stream>>>(mpool, gw, gb, gates);
  // K5: combine + LayerNorm + silu(z)
  k5_combine_ln<<<BL, 512, 0, stream>>>(ys, gates, xz, onw, onb, yact);
  // K6: 1024 M-tiles x 3 column-groups
  k6_out_proj<<<(1024 * 3) / 8, 256, 0, stream>>>(yact, opw, out);
}